// Attention_82867099009484
// MI455X (gfx1250) — compile-verified
//
#include <hip/hip_runtime.h>
#include <math.h>

#define B_ 16
#define S_ 4096
#define H_ 1024
#define Q_ 1024
#define K_ 2048
#define SCHUNK 256
#define NSCHUNK (S_ / SCHUNK)   // 16 chunks over S for the context partials

typedef float v2f __attribute__((ext_vector_type(2)));
typedef float v8f __attribute__((ext_vector_type(8)));

// Native CDNA5 V_TANH_F32 (quarter-rate trans op) if the toolchain exposes it;
// fall back to OCML tanhf otherwise (compile-safe either way).
#if __has_builtin(__builtin_amdgcn_tanh_f32)
#define TANHF(x) __builtin_amdgcn_tanh_f32(x)
#elif __has_builtin(__builtin_amdgcn_tanhf)
#define TANHF(x) __builtin_amdgcn_tanhf(x)
#else
#define TANHF(x) tanhf(x)
#endif

// ---------------------------------------------------------------------------
// K1: q[b][h] = sum_q query[b][q] * Wq[h][q]
// GEMM M=16 (batch), N=H_, K=Q_  -> exact-f32 via V_WMMA_F32_16X16X4_F32.
// One wave per 16x16 output tile; 64 tiles over N.
// ---------------------------------------------------------------------------
__global__ __launch_bounds__(32) void qproj_wmma(const float* __restrict__ query,
                                                 const float* __restrict__ Wq,
                                                 float* __restrict__ qout) {
  const int n0   = blockIdx.x * 16;
  const int lane = threadIdx.x;       // wave32: 0..31, EXEC all ones
  const int half = lane >> 4;         // 0 -> K pair {0,1}, 1 -> K pair {2,3}
  const int l15  = lane & 15;

  v8f c = {};
  for (int k0 = 0; k0 < Q_; k0 += 4) {
    const int ka = k0 + half * 2;
    v2f a, b;
    // A (16x4, M x K): lane row M=l15
    a.x = query[l15 * Q_ + ka];
    a.y = query[l15 * Q_ + ka + 1];
    // B (4x16, K x N): B[k][n] = Wq[n0+l15][k]  (Wq is H x Q row-major)
    b.x = Wq[(size_t)(n0 + l15) * Q_ + ka];
    b.y = Wq[(size_t)(n0 + l15) * Q_ + ka + 1];
    c = __builtin_amdgcn_wmma_f32_16x16x4_f32(false, a, false, b,
                                              (short)0, c, false, false);
  }
  // D 16x16 f32: lane column n = l15; VGPR r -> row (half*8 + r)
  for (int r = 0; r < 8; ++r) {
    const int m = half * 8 + r;       // batch index
    qout[m * H_ + n0 + l15] = c[r];
  }
}

// ---------------------------------------------------------------------------
// K2: scores[b][s] = sum_h tanh(q[b][h] + proj_key[b][s][h]) * w_energy[h]
// then apply mask (-inf where mask==0). One wave per (b,s) row, 8 rows/block.
// Streams proj_key (256 MB) with float4 loads; q-row + w_energy in LDS
// (read back as float4 -> ds_load_b128). tanh via native V_TANH_F32.
// ---------------------------------------------------------------------------
__global__ __launch_bounds__(256) void scores_kernel(const float* __restrict__ qout,
                                                     const float* __restrict__ proj_key,
                                                     const float* __restrict__ w_energy,
                                                     const int*   __restrict__ mask,
                                                     float* __restrict__ scores_out) {
  __shared__ float qs[H_];
  __shared__ float ws[H_];
  const int b  = blockIdx.x / (S_ / 8);
  const int s0 = (blockIdx.x % (S_ / 8)) * 8;

  for (int i = threadIdx.x; i < H_; i += 256) {
    qs[i] = qout[b * H_ + i];
    ws[i] = w_energy[i];
  }
  __syncthreads();

  const int wave = threadIdx.x >> 5;
  const int lane = threadIdx.x & 31;
  const int s    = s0 + wave;
  const float4* pk  = (const float4*)(proj_key + ((size_t)b * S_ + s) * H_);
  const float4* qs4 = (const float4*)qs;
  const float4* ws4 = (const float4*)ws;

  float acc = 0.f;
#pragma unroll
  for (int it = 0; it < H_ / 128; ++it) {
    const int h4 = it * 32 + lane;    // float4 index within the row (coalesced)
    const float4 v = pk[h4];
    const float4 q = qs4[h4];         // ds_load_b128
    const float4 w = ws4[h4];         // ds_load_b128
    acc += TANHF(q.x + v.x) * w.x;
    acc += TANHF(q.y + v.y) * w.y;
    acc += TANHF(q.z + v.z) * w.z;
    acc += TANHF(q.w + v.w) * w.w;
  }
  // wave32 reduction
  for (int off = 16; off > 0; off >>= 1) acc += __shfl_down(acc, off, 32);
  if (lane == 0) {
    scores_out[b * S_ + s] = (mask[b * S_ + s] == 0) ? -INFINITY : acc;
  }
}

// ---------------------------------------------------------------------------
// K3: in-place masked softmax over each batch row of 4096 scores.
// One block per batch; deterministic LDS tree reductions.
// ---------------------------------------------------------------------------
__global__ __launch_bounds__(256) void softmax_kernel(float* __restrict__ alphas) {
  __shared__ float red[256];
  float* row = alphas + (size_t)blockIdx.x * S_;

  float m = -INFINITY;
  for (int i = threadIdx.x; i < S_; i += 256) m = fmaxf(m, row[i]);
  red[threadIdx.x] = m; __syncthreads();
  for (int st = 128; st > 0; st >>= 1) {
    if (threadIdx.x < st) red[threadIdx.x] = fmaxf(red[threadIdx.x], red[threadIdx.x + st]);
    __syncthreads();
  }
  m = red[0]; __syncthreads();

  float sum = 0.f;
  for (int i = threadIdx.x; i < S_; i += 256) {
    const float e = expf(row[i] - m);   // exp(-inf - m) = 0 for masked lanes
    row[i] = e;
    sum += e;
  }
  red[threadIdx.x] = sum; __syncthreads();
  for (int st = 128; st > 0; st >>= 1) {
    if (threadIdx.x < st) red[threadIdx.x] += red[threadIdx.x + st];
    __syncthreads();
  }
  const float inv = 1.0f / red[0];
  for (int i = threadIdx.x; i < S_; i += 256) row[i] *= inv;
}

// ---------------------------------------------------------------------------
// K4: context partials. Splits S into 16 chunks for parallelism (512 blocks),
// streams value (512 MB) with float4 loads; alphas chunk cached in LDS.
// ---------------------------------------------------------------------------
__global__ __launch_bounds__(256) void context_partial(const float* __restrict__ alphas,
                                                       const float* __restrict__ value,
                                                       float* __restrict__ partials) {
  const int kchunks = K_ / 1024;      // 2 (256 threads * float4)
  int id = blockIdx.x;
  const int kc = id % kchunks; id /= kchunks;
  const int sc = id % NSCHUNK; id /= NSCHUNK;
  const int b  = id;

  __shared__ float al[SCHUNK];
  for (int i = threadIdx.x; i < SCHUNK; i += 256)
    al[i] = alphas[b * S_ + sc * SCHUNK + i];
  __syncthreads();

  const int k = kc * 1024 + threadIdx.x * 4;
  const float4* vp = (const float4*)(value + ((size_t)b * S_ + (size_t)sc * SCHUNK) * K_ + k);

  float4 acc = {0.f, 0.f, 0.f, 0.f};
  for (int s = 0; s < SCHUNK; ++s) {
    const float a  = al[s];
    const float4 v = vp[(size_t)s * (K_ / 4)];
    acc.x += a * v.x; acc.y += a * v.y; acc.z += a * v.z; acc.w += a * v.w;
  }
  *(float4*)(partials + ((size_t)(b * NSCHUNK + sc)) * K_ + k) = acc;
}

// ---------------------------------------------------------------------------
// K5: deterministic reduction of the 16 S-chunk partials into context.
// ---------------------------------------------------------------------------
__global__ __launch_bounds__(256) void context_reduce(const float* __restrict__ partials,
                                                      float* __restrict__ ctx) {
  const int idx = blockIdx.x * 256 + threadIdx.x;   // over B_*K_
  const int b = idx / K_;
  const int k = idx % K_;
  float s = 0.f;
#pragma unroll
  for (int c = 0; c < NSCHUNK; ++c)
    s += partials[((size_t)(b * NSCHUNK + c)) * K_ + k];
  ctx[idx] = s;
}

// ---------------------------------------------------------------------------
extern "C" void kernel_launch(void* const* d_in, const int* in_sizes, int n_in,
                              void* d_out, int out_size, void* d_ws, size_t ws_size,
                              hipStream_t stream) {
  const int*   mask     = (const int*)  d_in[0];   // (B,1,S)
  const float* query    = (const float*)d_in[1];   // (B,1,Q)
  const float* proj_key = (const float*)d_in[2];   // (B,S,H)
  const float* value    = (const float*)d_in[3];   // (B,S,K)
  const float* Wq       = (const float*)d_in[4];   // (H,Q)
  const float* w_energy = (const float*)d_in[5];   // (H,)

  float* ctx_out   = (float*)d_out;                // context: B_*K_ floats
  float* alpha_out = (float*)d_out + (size_t)B_ * K_;  // alphas: B_*S_ floats

  float* q_ws      = (float*)d_ws;                          // B_*H_ floats
  float* partials  = (float*)d_ws + (size_t)B_ * H_;        // B_*NSCHUNK*K_ floats

  // 1) q = query @ Wq^T  (WMMA f32 16x16x4; 64 tiles over H)
  qproj_wmma<<<H_ / 16, 32, 0, stream>>>(query, Wq, q_ws);

  // 2) masked scores -> alpha region of d_out
  scores_kernel<<<B_ * S_ / 8, 256, 0, stream>>>(q_ws, proj_key, w_energy, mask, alpha_out);

  // 3) softmax in place
  softmax_kernel<<<B_, 256, 0, stream>>>(alpha_out);

  // 4) context partials over 16 S-chunks
  context_partial<<<B_ * NSCHUNK * (K_ / 1024), 256, 0, stream>>>(alpha_out, value, partials);

  // 5) deterministic final reduce
  context_reduce<<<B_ * K_ / 256, 256, 0, stream>>>(partials, ctx_out);
}